// CustomPoolingLayer_30734785970690
// MI455X (gfx1250) — compile-verified
//
#include <hip/hip_runtime.h>
#include <stdint.h>

// Problem constants (from reference)
#define B_  16
#define C_  64
#define H_  224
#define W_  224
#define WIN 7
#define HO  (H_ - WIN + 1)          // 218
#define WO  (W_ - WIN + 1)          // 218
#define NPIX (HO * WO)              // 47524
#define BC  (B_ * C_)               // 1024
#define QDIV 7.0f                   // WIN - R + 1 with R=1

// Tiling
#define TH 16                        // output rows per tile
#define RAW_ROWS (TH + WIN - 1)      // 22 input rows per tile
#define W4 (W_ / 4)                  // 56 float4 per row
#define BC_CHUNK 16                  // (b,c) planes per block
#define NCHUNK (BC / BC_CHUNK)       // 64
#define NTILE ((HO + TH - 1) / TH)   // 14

// ---- CDNA5 async global->LDS path (ASYNCcnt) ------------------------------
#if defined(__gfx1250__) && __has_builtin(__builtin_amdgcn_global_load_async_to_lds_b32)
#define HAVE_ASYNC_LDS 1
#endif
#if defined(__gfx1250__) && __has_builtin(__builtin_amdgcn_global_load_async_to_lds_b128)
#define HAVE_ASYNC_B128 1
#endif

typedef __attribute__((ext_vector_type(4))) int v4i;
typedef __attribute__((address_space(1))) int gint_t;
typedef __attribute__((address_space(3))) int lint_t;
typedef __attribute__((address_space(1))) v4i gv4i_t;
typedef __attribute__((address_space(3))) v4i lv4i_t;

__device__ __forceinline__ void async_g2l_b32(const float* g, float* l) {
#ifdef HAVE_ASYNC_LDS
  __builtin_amdgcn_global_load_async_to_lds_b32(
      (gint_t*)(uintptr_t)g,
      (lint_t*)(uintptr_t)(uint32_t)(uintptr_t)l,
      /*offset=*/0, /*cpol=*/0);
#else
  *l = *g;
#endif
}

#ifdef HAVE_ASYNC_B128
__device__ __forceinline__ void async_g2l_b128(const float* g, float* l) {
  __builtin_amdgcn_global_load_async_to_lds_b128(
      (gv4i_t*)(uintptr_t)g,
      (lv4i_t*)(uintptr_t)(uint32_t)(uintptr_t)l,
      /*offset=*/0, /*cpol=*/0);
}
#endif

__device__ __forceinline__ void wait_async0() {
#ifdef HAVE_ASYNC_LDS
#if __has_builtin(__builtin_amdgcn_s_wait_asynccnt)
  __builtin_amdgcn_s_wait_asynccnt(0);
#else
  asm volatile("s_wait_asynccnt 0" ::: "memory");
#endif
  asm volatile("" ::: "memory");   // keep LDS reads after the wait
#endif
}

// ---- Core tile computation -------------------------------------------------
// For one (b,c) plane and one 16-row tile: async-copy the contiguous
// nrows x 224 slab into LDS (B128 flat copy when available), then vertical
// 7-tap min/max per column, then horizontal 7-tap min/max; emit
// nr = ceil(max) - ceil(min) - 1 for each of the TH output rows.
template <typename F>
__device__ __forceinline__ void tile_nr(const float* __restrict__ plane,
                                        int ho0, int nrows,
                                        float (*raw)[W_],
                                        float (*vmx)[W_], float (*vmn)[W_],
                                        int x, F&& emit) {
#ifdef HAVE_ASYNC_B128
  {
    // Slab is contiguous: rows ho0..ho0+nrows-1, full width.
    const float* src = plane + (size_t)ho0 * W_;
    float* dst = &raw[0][0];
    const int nf4 = nrows * W4;                       // float4 chunks (<=1232)
#pragma unroll
    for (int i = 0; i < (RAW_ROWS * W4 + W_ - 1) / W_; ++i) {   // 6 iters
      int f = i * W_ + x;
      if (f < nf4) async_g2l_b128(src + 4 * (size_t)f, dst + 4 * (size_t)f);
    }
  }
#else
#pragma unroll
  for (int j = 0; j < RAW_ROWS; ++j) {
    if (j < nrows) async_g2l_b32(plane + (size_t)(ho0 + j) * W_ + x, &raw[j][x]);
  }
#endif
  wait_async0();
  __syncthreads();                         // raw visible to all columns

#pragma unroll
  for (int t = 0; t < TH; ++t) {          // vertical window over own column
    float a = raw[t][x], b = a;
#pragma unroll
    for (int k = 1; k < WIN; ++k) {
      float v = raw[t + k][x];
      a = fmaxf(a, v);
      b = fminf(b, v);
    }
    vmx[t][x] = a;
    vmn[t][x] = b;
  }
  __syncthreads();                         // vmx/vmn visible to neighbors

  if (x < WO) {
#pragma unroll
    for (int t = 0; t < TH; ++t) {        // horizontal window
      float a = vmx[t][x], b = vmn[t][x];
#pragma unroll
      for (int k = 1; k < WIN; ++k) {
        a = fmaxf(a, vmx[t][x + k]);
        b = fminf(b, vmn[t][x + k]);
      }
      emit(t, __builtin_ceilf(a) - __builtin_ceilf(b) - 1.0f);
    }
  }
  __syncthreads();                         // protect raw/vmx/vmn for next plane
}

// ---- Pass 1: per-chunk partial sums of nr over (b,c) -----------------------
template <bool USE_ATOMIC>
__global__ __launch_bounds__(W_) void pool_pass1(const float* __restrict__ img,
                                                 float* __restrict__ dst) {
  __shared__ __align__(16) float raw[RAW_ROWS][W_];
  __shared__ float vmx[TH][W_];
  __shared__ float vmn[TH][W_];

  const int x = threadIdx.x;                    // 0..223
  const int ho0 = blockIdx.y * TH;
  const int nho = (HO - ho0 < TH) ? (HO - ho0) : TH;
  const int nrows = nho + WIN - 1;
  const int bc0 = blockIdx.x * BC_CHUNK;

  float acc[TH];
#pragma unroll
  for (int t = 0; t < TH; ++t) acc[t] = 0.0f;

  for (int c = 0; c < BC_CHUNK; ++c) {
    const float* plane = img + (size_t)(bc0 + c) * (H_ * W_);
    tile_nr(plane, ho0, nrows, raw, vmx, vmn, x,
            [&](int t, float nr) { acc[t] += nr; });
  }

  if (x < WO) {
    for (int t = 0; t < nho; ++t) {
      if (USE_ATOMIC) {
        atomicAdd(&dst[(ho0 + t) * WO + x], acc[t]);
      } else {
        dst[(size_t)blockIdx.x * NPIX + (ho0 + t) * WO + x] = acc[t];
      }
    }
  }
}

// ---- Reduce 64 partial maps into Mr (fixed order => deterministic) ---------
__global__ void reduce_partials(const float* __restrict__ part,
                                float* __restrict__ Mr) {
  int i = blockIdx.x * 256 + threadIdx.x;
  if (i < NPIX) {
    float s = 0.0f;
    for (int c = 0; c < NCHUNK; ++c) s += part[(size_t)c * NPIX + i];
    Mr[i] = s;
  }
}

__global__ void zero_f32(float* p, int n) {
  int i = blockIdx.x * 256 + threadIdx.x;
  if (i < n) p[i] = 0.0f;
}

// ---- Pass 2: recompute nr, apply faithful L_r formula, store output --------
__global__ __launch_bounds__(W_) void pool_pass2(const float* __restrict__ img,
                                                 const float* __restrict__ Mr,
                                                 float* __restrict__ out) {
  __shared__ __align__(16) float raw[RAW_ROWS][W_];
  __shared__ float vmx[TH][W_];
  __shared__ float vmn[TH][W_];

  const int x = threadIdx.x;
  const int ho0 = blockIdx.y * TH;
  const int nho = (HO - ho0 < TH) ? (HO - ho0) : TH;
  const int nrows = nho + WIN - 1;
  const int bc0 = blockIdx.x * BC_CHUNK;

  float m[TH];
#pragma unroll
  for (int t = 0; t < TH; ++t) m[t] = 0.0f;
  if (x < WO) {
    for (int t = 0; t < nho; ++t) m[t] = Mr[(ho0 + t) * WO + x];
  }

  for (int c = 0; c < BC_CHUNK; ++c) {
    const float* plane = img + (size_t)(bc0 + c) * (H_ * W_);
    const size_t obase = (size_t)(bc0 + c) * NPIX;
    tile_nr(plane, ho0, nrows, raw, vmx, vmn, x, [&](int t, float nr) {
      if (t < nho) {
        // Faithful to reference: L_r = Mr^2 * Q / (Mr * Q)^2, Q = nr / 7
        float q   = nr / QDIV;
        float mm  = m[t];
        float num = mm * mm * q;
        float den = (mm * q) * (mm * q);
        __builtin_nontemporal_store(num / den,
                                    &out[obase + (size_t)(ho0 + t) * WO + x]);
      }
    });
  }
}

// ---- Launcher --------------------------------------------------------------
extern "C" void kernel_launch(void* const* d_in, const int* in_sizes, int n_in,
                              void* d_out, int out_size, void* d_ws, size_t ws_size,
                              hipStream_t stream) {
  const float* img = (const float*)d_in[0];
  float* out = (float*)d_out;
  float* Mr   = (float*)d_ws;            // NPIX floats
  float* part = Mr + NPIX;               // NCHUNK * NPIX floats

  dim3 grid(NCHUNK, NTILE);              // 64 x 14 blocks, 224 threads each
  const size_t need_det = (size_t)(NCHUNK + 1) * NPIX * sizeof(float);

  if (ws_size >= need_det) {
    // Deterministic path: per-chunk partials + fixed-order reduction.
    pool_pass1<false><<<grid, W_, 0, stream>>>(img, part);
    reduce_partials<<<(NPIX + 255) / 256, 256, 0, stream>>>(part, Mr);
  } else {
    // Fallback: float atomics into Mr (zeroed every call).
    zero_f32<<<(NPIX + 255) / 256, 256, 0, stream>>>(Mr, NPIX);
    pool_pass1<true><<<grid, W_, 0, stream>>>(img, Mr);
  }
  pool_pass2<<<grid, W_, 0, stream>>>(img, Mr, out);
}